// FlowDecoderLayer_26139170964333
// MI455X (gfx1250) — compile-verified
//
#include <hip/hip_runtime.h>
#include <math.h>

typedef float v2f __attribute__((ext_vector_type(2)));
typedef float v4f __attribute__((ext_vector_type(4)));
typedef float v8f __attribute__((ext_vector_type(8)));

namespace {
constexpr int kBS     = 4096;
constexpr int kDM     = 256;   // flow embed dim
constexpr int kDInner = 2048;
constexpr int kDtRank = 16;
constexpr int kDState = 16;
constexpr int kProjW  = 48;    // dt_rank + 2*d_state
}

__device__ __forceinline__ float silu_f(float x) {
  return x / (1.0f + __expf(-x));
}
__device__ __forceinline__ float softplus_f(float x) {
  // stable log(1 + exp(x)) = max(x,0) + log1p(exp(-|x|))
  return fmaxf(x, 0.0f) + log1pf(__expf(-fabsf(x)));
}

// ---------------------------------------------------------------------------
// Kernel 0: Aneg[d,n] = -exp(A_log[d,n]).  2048x16 = 32768 elems, one time.
// The table (128 KB) then stays L2-resident and removes 134M v_exp_f32 from
// the streaming kernel's hot loop (keeps trans work below the HBM roofline).
// ---------------------------------------------------------------------------
__global__ void __launch_bounds__(256) k_negexp_A(const float* __restrict__ A_log,
                                                  float* __restrict__ Aneg) {
  const int i = blockIdx.x * blockDim.x + threadIdx.x;
  Aneg[i] = -expf(A_log[i]);
}

// ---------------------------------------------------------------------------
// Kernel 1: proj = flow_embed (4096x256) @ W_flow.T (256x48) using
// V_WMMA_F32_16X16X4_F32. One wave per 16x16 output tile, 64 K=4 steps.
// A layout (16x4 f32): lanes 0-15 row M=lane, v0=K0 v1=K1; lanes 16-31 K2/K3.
// B layout (4x16 f32): symmetric (B[k][n] = W_flow[n][k]).
// C/D layout: VGPR r -> row r + 8*hi, col = lane&15.
// ---------------------------------------------------------------------------
__global__ void __launch_bounds__(128) k_proj_wmma(const float* __restrict__ flow,
                                                   const float* __restrict__ wflow,
                                                   float* __restrict__ proj) {
  const int wave = (blockIdx.x * blockDim.x + threadIdx.x) >> 5;  // 0..767
  const int lane = threadIdx.x & 31;
  const int col  = lane & 15;
  const int hi   = lane >> 4;
  const int mt = wave / 3;        // 0..255 (batch tile)
  const int nt = wave - mt * 3;   // 0..2   (output-col tile, 3*16 = 48)

  const float* ap = flow  + (size_t)(mt * 16 + col) * kDM + 2 * hi;
  const float* bp = wflow + (size_t)(nt * 16 + col) * kDM + 2 * hi;

  v8f acc = {};
  #pragma unroll 4
  for (int kc = 0; kc < kDM / 4; ++kc) {
    v2f a, b;
    a.x = ap[0]; a.y = ap[1];
    b.x = bp[0]; b.y = bp[1];
    ap += 4; bp += 4;
    acc = __builtin_amdgcn_wmma_f32_16x16x4_f32(false, a, false, b,
                                                (short)0, acc, false, false);
  }

  float* op = proj + (size_t)(mt * 16 + 8 * hi) * kProjW + nt * 16 + col;
  #pragma unroll
  for (int r = 0; r < 8; ++r) op[(size_t)r * kProjW] = acc[r];
}

// ---------------------------------------------------------------------------
// Kernel 2: fused selective state update. One thread per (b, d) pair.
// Computes dt / e / silu(r) inline (operands are L2-resident), streams the
// 536 MB hidden state in/out with non-temporal b128 accesses, writes the
// per-(b,d) scalar y into workspace.
// ---------------------------------------------------------------------------
__global__ void __launch_bounds__(256) k_state_update(
    const float* __restrict__ bbox,  const float* __restrict__ h_in,
    const float* __restrict__ proj,  const float* __restrict__ W_in,
    const float* __restrict__ b_in,  const float* __restrict__ W_dt,
    const float* __restrict__ b_dt,  const float* __restrict__ Aneg,
    const float* __restrict__ Dvec,  float* __restrict__ h_out,
    float* __restrict__ ybuf) {
  const int tid = blockIdx.x * blockDim.x + threadIdx.x;  // 0 .. BS*D_INNER-1
  const int b = tid >> 11;               // / D_INNER
  const int d = tid & (kDInner - 1);
  const float* pr = proj + (size_t)b * kProjW;

  // dt = softplus(proj[b,0:16] . W_dt[d,:] + b_dt[d])
  const v4f* wdt4 = reinterpret_cast<const v4f*>(W_dt + (size_t)d * kDtRank);
  float dtacc = b_dt[d];
  #pragma unroll
  for (int q = 0; q < 4; ++q) {
    const v4f w = wdt4[q];
    dtacc += pr[4*q+0]*w.x + pr[4*q+1]*w.y + pr[4*q+2]*w.z + pr[4*q+3]*w.w;
  }
  const float dt = softplus_f(dtacc);

  // e = silu(bbox[b] . W_in[d] + b_in[d]); rs = silu(bbox[b] . W_in[d+2048] + ...)
  const v4f bb = *reinterpret_cast<const v4f*>(bbox + (size_t)b * 4);
  const v4f we = *reinterpret_cast<const v4f*>(W_in + (size_t)d * 4);
  const v4f wr = *reinterpret_cast<const v4f*>(W_in + (size_t)(d + kDInner) * 4);
  const float e  = silu_f(b_in[d]           + bb.x*we.x + bb.y*we.y + bb.z*we.z + bb.w*we.w);
  const float rs = silu_f(b_in[d + kDInner] + bb.x*wr.x + bb.y*wr.y + bb.z*wr.z + bb.w*wr.w);

  const float edt = e * dt;
  const v4f* an4 = reinterpret_cast<const v4f*>(Aneg + (size_t)d * kDState);
  const float* Bv = pr + kDtRank;            // proj[b, 16:32]
  const float* Cv = pr + kDtRank + kDState;  // proj[b, 32:48]

  const v4f* hp = reinterpret_cast<const v4f*>(h_in + (size_t)tid * kDState);
  v4f* ho = reinterpret_cast<v4f*>(h_out + (size_t)tid * kDState);

  float y = 0.0f;
  #pragma unroll
  for (int q = 0; q < 4; ++q) {
    const v4f hv = __builtin_nontemporal_load(hp + q);
    const v4f av = an4[q];
    const float h4[4] = {hv.x, hv.y, hv.z, hv.w};
    const float a4[4] = {av.x, av.y, av.z, av.w};
    float nh[4];
    #pragma unroll
    for (int j = 0; j < 4; ++j) {
      const int n = 4*q + j;
      const float dA = __expf(dt * a4[j]);    // exp(dt * A), A = -exp(A_log)
      const float hn = h4[j] * dA + edt * Bv[n];
      nh[j] = hn;
      y += hn * Cv[n];
    }
    v4f o; o.x = nh[0]; o.y = nh[1]; o.z = nh[2]; o.w = nh[3];
    __builtin_nontemporal_store(o, ho + q);
  }
  y = (y + Dvec[d] * e) * rs;
  ybuf[tid] = y;
}

// ---------------------------------------------------------------------------
// Kernel 3: out = y (4096x2048) @ W_out.T (2048x4) + b_out.
// One wave32 per batch row; lanes stride over d, then shuffle-reduce.
// ---------------------------------------------------------------------------
__global__ void __launch_bounds__(256) k_out_proj(const float* __restrict__ ybuf,
                                                  const float* __restrict__ W_out,
                                                  const float* __restrict__ b_out,
                                                  float* __restrict__ out) {
  const int row  = (blockIdx.x * blockDim.x + threadIdx.x) >> 5;  // batch index
  const int lane = threadIdx.x & 31;
  const float* yr = ybuf + (size_t)row * kDInner;
  float a0 = 0.f, a1 = 0.f, a2 = 0.f, a3 = 0.f;
  for (int d = lane; d < kDInner; d += 32) {
    const float v = yr[d];
    a0 += v * W_out[d];
    a1 += v * W_out[kDInner + d];
    a2 += v * W_out[2 * kDInner + d];
    a3 += v * W_out[3 * kDInner + d];
  }
  #pragma unroll
  for (int off = 16; off > 0; off >>= 1) {
    a0 += __shfl_down(a0, off, 32);
    a1 += __shfl_down(a1, off, 32);
    a2 += __shfl_down(a2, off, 32);
    a3 += __shfl_down(a3, off, 32);
  }
  if (lane == 0) {
    out[row * 4 + 0] = a0 + b_out[0];
    out[row * 4 + 1] = a1 + b_out[1];
    out[row * 4 + 2] = a2 + b_out[2];
    out[row * 4 + 3] = a3 + b_out[3];
  }
}

extern "C" void kernel_launch(void* const* d_in, const int* in_sizes, int n_in,
                              void* d_out, int out_size, void* d_ws, size_t ws_size,
                              hipStream_t stream) {
  (void)in_sizes; (void)n_in; (void)out_size; (void)ws_size;
  const float* bbox   = (const float*)d_in[0];   // (4096, 4)
  const float* h_in   = (const float*)d_in[1];   // (4096, 2048, 16)
  const float* flow   = (const float*)d_in[2];   // (4096, 256)
  const float* W_in   = (const float*)d_in[3];   // (4096, 4)
  const float* b_in   = (const float*)d_in[4];   // (4096,)
  const float* W_dt   = (const float*)d_in[5];   // (2048, 16)
  const float* b_dt   = (const float*)d_in[6];   // (2048,)
  const float* W_flow = (const float*)d_in[7];   // (48, 256)
  const float* A_log  = (const float*)d_in[8];   // (2048, 16)
  const float* Dvec   = (const float*)d_in[9];   // (2048,)
  const float* W_out  = (const float*)d_in[10];  // (4, 2048)
  const float* b_out  = (const float*)d_in[11];  // (4,)

  float* out   = (float*)d_out;                       // (4096, 4)
  float* h_out = out + (size_t)kBS * 4;               // (4096, 2048, 16)

  float* proj = (float*)d_ws;                         // (4096, 48)
  float* ybuf = proj + (size_t)kBS * kProjW;          // (4096, 2048)
  float* Aneg = ybuf + (size_t)kBS * kDInner;         // (2048, 16)

  // 0) Aneg = -exp(A_log): one-time 128 KB table (stays L2-resident)
  k_negexp_A<<<(kDInner * kDState) / 256, 256, 0, stream>>>(A_log, Aneg);

  // 1) flow projection via f32 WMMA: 256 M-tiles x 3 N-tiles = 768 waves
  k_proj_wmma<<<192, 128, 0, stream>>>(flow, W_flow, proj);

  // 2) fused selective state update: one thread per (b, d)
  k_state_update<<<(kBS * kDInner) / 256, 256, 0, stream>>>(
      bbox, h_in, proj, W_in, b_in, W_dt, b_dt, Aneg, Dvec, h_out, ybuf);

  // 3) output projection: one wave per batch row
  k_out_proj<<<kBS / 8, 256, 0, stream>>>(ybuf, W_out, b_out, out);
}